// Tramba_60507499266496
// MI455X (gfx1250) — compile-verified
//
#include <hip/hip_runtime.h>
#include <hip/hip_bf16.h>
#include <math.h>

// MI455X / gfx1250 implementation. wave32. fp32 GEMMs run on the matrix core
// via V_WMMA_F32_16X16X4_F32 (exact fp32, matches the fp32 reference).
//
// The adaptive-attention branch is mathematically dead for the final output:
// all shifts are strictly negative, so at t = T-1 every shifted tensor is
// zero-padded to exactly zero => attention output at the last timestep is 0
// => attn_last == layer_norm(0) == ln1_b (broadcast). We use that identity.

typedef float v2f __attribute__((ext_vector_type(2)));
typedef float v8f __attribute__((ext_vector_type(8)));

#define BB    2
#define TT    64
#define LL    256
#define DM    32
#define INF_  5
#define PREDN 12
#define DI    64
#define DSN   16
#define NSEQ  (BB * LL)     // 512 sequences
#define NROWS (NSEQ * TT)   // 32768 (row = n*TT + t, n = b*LL + l)

// ---------------------------------------------------------------------------
// K1: h = (x @ embed_w^T + embed_b + pos_emb[t]) @ agg_w^T + agg_b,
// written directly in mamba layout hm[(b*L+l)*T + t][d].
// ---------------------------------------------------------------------------
__global__ __launch_bounds__(256) void k_embed(
    const float* __restrict__ x, const float* __restrict__ embw,
    const float* __restrict__ embb, const float* __restrict__ pos,
    const float* __restrict__ aggw, const float* __restrict__ aggb,
    float* __restrict__ hm)
{
    __shared__ float s_ew[DM * INF_];
    __shared__ float s_eb[DM];
    __shared__ float s_aw[DM * DM];
    __shared__ float s_ab[DM];
    __shared__ float s_pos[TT * DM];
    for (int i = threadIdx.x; i < DM * INF_; i += 256) s_ew[i] = embw[i];
    for (int i = threadIdx.x; i < DM; i += 256) { s_eb[i] = embb[i]; s_ab[i] = aggb[i]; }
    for (int i = threadIdx.x; i < DM * DM; i += 256) s_aw[i] = aggw[i];
    for (int i = threadIdx.x; i < TT * DM; i += 256) s_pos[i] = pos[i];
    __syncthreads();

    int idx = blockIdx.x * 256 + threadIdx.x;      // flat (b,t,l)
    if (idx >= BB * TT * LL) return;
    int l  = idx % LL;
    int bt = idx / LL;
    int t  = bt % TT;
    int b  = bt / TT;

    float xin[INF_];
    const float* xp = x + (size_t)idx * INF_;
#pragma unroll
    for (int f = 0; f < INF_; ++f) xin[f] = xp[f];

    float e[DM];
#pragma unroll
    for (int d = 0; d < DM; ++d) {
        float a = s_eb[d];
#pragma unroll
        for (int f = 0; f < INF_; ++f) a += s_ew[d * INF_ + f] * xin[f];
        e[d] = a + s_pos[t * DM + d];
    }
    size_t orow = ((size_t)(b * LL + l) * TT + t) * DM;
    for (int d = 0; d < DM; ++d) {
        float a = s_ab[d];
#pragma unroll
        for (int j = 0; j < DM; ++j) a += s_aw[d * DM + j] * e[j];
        hm[orow + d] = a;
    }
}

// ---------------------------------------------------------------------------
// K2: RMSNorm over D=32 (one wave per row, lane == channel).
// ---------------------------------------------------------------------------
__global__ __launch_bounds__(256) void k_rms(
    const float* __restrict__ hm, const float* __restrict__ w,
    float* __restrict__ u, int rows)
{
    int wid  = blockIdx.x * 8 + (threadIdx.x >> 5);
    int lane = threadIdx.x & 31;
    if (wid >= rows) return;
    float v = hm[(size_t)wid * DM + lane];
    float ss = v * v;
    for (int off = 16; off >= 1; off >>= 1) ss += __shfl_xor(ss, off, 32);
    float r = rsqrtf(ss * (1.0f / DM) + 1e-5f);
    u[(size_t)wid * DM + lane] = v * r * w[lane];
}

// ---------------------------------------------------------------------------
// WMMA NT-GEMM: C[M,N] = A[M,K] * B[N,K]^T (+bias) (+res) (relu optional).
// One 16x16 tile per wave using V_WMMA_F32_16X16X4_F32. Templated on K
// (multiple of 4) so the loop fully unrolls into a load batch + WMMA chain.
// Requirements: M % 16 == 0; all row strides even (8B-aligned v2f loads).
// Ragged N (N % 16 != 0): out-of-range B columns are clamped to column N-1;
// they pollute only output columns that are never stored (no cross-column
// mixing in GEMM), so EXEC stays all-1s through the WMMA loop (ISA 7.12).
//
// Fragment layout (ISA 7.12.2, 32-bit 16x4 A): lane&15 = row/col index,
// lane>>4 selects K pair {0,1} vs {2,3}; C/D: vgpr r -> M = r + 8*(lane>>4),
// N = lane&15.  res/C may alias (in-place residual add) => not __restrict__.
// ---------------------------------------------------------------------------
template <int K>
__global__ __launch_bounds__(128) void k_gemm_nt(
    const float* __restrict__ A, int lda,
    const float* __restrict__ Bw,          // N x K row-major (weights)
    const float* __restrict__ bias,        // N or nullptr
    const float* res, int ldres,           // M x N (strided) or nullptr
    float* C, int ldc,
    int M, int N, int ntn, int relu)
{
    int wave = blockIdx.x * 4 + (threadIdx.x >> 5);
    int lane = threadIdx.x & 31;
    int mt   = wave / ntn;
    int nt   = wave - mt * ntn;
    if (mt * 16 >= M) return;                    // wave-uniform exit

    int half2 = (lane >> 4) << 1;                // 0 or 2
    int l15   = lane & 15;
    int bcol  = nt * 16 + l15;
    int bcolc = bcol < N ? bcol : (N - 1);       // clamp (see header comment)
    const float* Ar = A + (size_t)(mt * 16 + l15) * lda + half2;
    const float* Br = Bw + (size_t)bcolc * K + half2;

    v8f acc = {};
#pragma unroll
    for (int k0 = 0; k0 < K; k0 += 4) {
        v2f a  = *(const v2f*)(Ar + k0);
        v2f bf = *(const v2f*)(Br + k0);
        acc = __builtin_amdgcn_wmma_f32_16x16x4_f32(
            false, a, false, bf, (short)0, acc, false, false);
    }

    if (bcol >= N) return;                       // diverge only after WMMA
    float bv = bias ? bias[bcol] : 0.0f;
    int mbase = mt * 16 + ((lane >> 4) << 3);
#pragma unroll
    for (int r = 0; r < 8; ++r) {
        int m = mbase + r;
        float v = acc[r] + bv;
        if (relu) v = fmaxf(v, 0.0f);
        if (res)  v += res[(size_t)m * ldres + bcol];
        C[(size_t)m * ldc + bcol] = v;
    }
}

// ---------------------------------------------------------------------------
// K4: causal depthwise conv (D_CONV=4) over time + SiLU. xi = xz[:, 0:64].
// ---------------------------------------------------------------------------
__global__ __launch_bounds__(256) void k_conv(
    const float* __restrict__ xz, const float* __restrict__ cw,
    const float* __restrict__ cb, float* __restrict__ xc)
{
    int idx = blockIdx.x * 256 + threadIdx.x;
    if (idx >= NROWS * DI) return;
    int d   = idx & 63;
    int row = idx >> 6;
    int t   = row & 63;
    int n   = row >> 6;
    float acc = cb[d];
#pragma unroll
    for (int k = 0; k < 4; ++k) {
        int tt2 = t + k - 3;
        if (tt2 >= 0) acc += xz[((size_t)(n * TT + tt2)) * (2 * DI) + d] * cw[d * 4 + k];
    }
    xc[idx] = acc / (1.0f + expf(-acc));          // silu
}

// ---------------------------------------------------------------------------
// K6: dt = softplus(dbl[:, 0:2] @ dt_w^T + dt_b)   (K=2: VALU is optimal)
// ---------------------------------------------------------------------------
__global__ __launch_bounds__(256) void k_dt(
    const float* __restrict__ dbl, const float* __restrict__ dtw,
    const float* __restrict__ dtbias, float* __restrict__ dtb)
{
    int idx = blockIdx.x * 256 + threadIdx.x;
    if (idx >= NROWS * DI) return;
    int d   = idx & 63;
    int row = idx >> 6;
    float s = dbl[(size_t)row * 34] * dtw[d * 2]
            + dbl[(size_t)row * 34 + 1] * dtw[d * 2 + 1] + dtbias[d];
    dtb[idx] = fmaxf(s, 0.0f) + log1pf(expf(-fabsf(s)));   // stable softplus
}

// ---------------------------------------------------------------------------
// K7: selective scan. One wave per sequence; lane owns channels d0=2*lane,
// d1=2*lane+1 with 16 states each in registers. B/C (32 values) loaded as one
// float per lane and broadcast via __shfl. Epilogue: +D*xc, *silu(z).
// ---------------------------------------------------------------------------
__global__ __launch_bounds__(128) void k_scan(
    const float* __restrict__ xc, const float* __restrict__ dbl,
    const float* __restrict__ dtb, const float* __restrict__ xz,
    const float* __restrict__ alog, const float* __restrict__ dvec,
    float* __restrict__ yb)
{
    int wave = blockIdx.x * 4 + (threadIdx.x >> 5);
    int lane = threadIdx.x & 31;
    if (wave >= NSEQ) return;
    int n  = wave;
    int d0 = lane * 2, d1 = d0 + 1;

    float A0[DSN], A1[DSN], h0[DSN], h1[DSN];
#pragma unroll
    for (int s = 0; s < DSN; ++s) {
        A0[s] = -expf(alog[d0 * DSN + s]);
        A1[s] = -expf(alog[d1 * DSN + s]);
        h0[s] = 0.0f; h1[s] = 0.0f;
    }
    float Dd0 = dvec[d0], Dd1 = dvec[d1];

#pragma unroll 1
    for (int t = 0; t < TT; ++t) {
        size_t row = (size_t)n * TT + t;
        float bc  = dbl[row * 34 + 2 + lane];     // lane<16: B[s], lane>=16: C[s-16]
        float dt0 = dtb[row * DI + d0], dt1 = dtb[row * DI + d1];
        float x0  = xc[row * DI + d0],  x1  = xc[row * DI + d1];
        float y0 = 0.0f, y1 = 0.0f;
#pragma unroll
        for (int s = 0; s < DSN; ++s) {
            float Bs = __shfl(bc, s, 32);
            float Cs = __shfl(bc, s + 16, 32);
            h0[s] = expf(dt0 * A0[s]) * h0[s] + dt0 * Bs * x0;
            h1[s] = expf(dt1 * A1[s]) * h1[s] + dt1 * Bs * x1;
            y0 += h0[s] * Cs;
            y1 += h1[s] * Cs;
        }
        y0 += Dd0 * x0;
        y1 += Dd1 * x1;
        float z0 = xz[row * (2 * DI) + DI + d0];
        float z1 = xz[row * (2 * DI) + DI + d1];
        y0 *= z0 / (1.0f + expf(-z0));            // y * silu(z)
        y1 *= z1 / (1.0f + expf(-z1));
        yb[row * DI + d0] = y0;
        yb[row * DI + d1] = y1;
    }
}

// ---------------------------------------------------------------------------
// K9: LayerNorm over D=32 (one wave per row).
// ---------------------------------------------------------------------------
__global__ __launch_bounds__(256) void k_ln(
    const float* __restrict__ in, const float* __restrict__ g,
    const float* __restrict__ b, float* __restrict__ out, int rows)
{
    int wid  = blockIdx.x * 8 + (threadIdx.x >> 5);
    int lane = threadIdx.x & 31;
    if (wid >= rows) return;
    float v = in[(size_t)wid * DM + lane];
    float s = v;
    for (int off = 16; off >= 1; off >>= 1) s += __shfl_xor(s, off, 32);
    float mean = s * (1.0f / DM);
    float dv = v - mean;
    float s2 = dv * dv;
    for (int off = 16; off >= 1; off >>= 1) s2 += __shfl_xor(s2, off, 32);
    float var = s2 * (1.0f / DM);
    out[(size_t)wid * DM + lane] = dv * rsqrtf(var + 1e-5f) * g[lane] + b[lane];
}

// ---------------------------------------------------------------------------
// K10: fused = sigmoid(gate)*ln1_b + (1-g)*mamba_last; y = fused@fc_w^T+fc_b,
// written transposed: out[b, p, l].
// ---------------------------------------------------------------------------
__global__ __launch_bounds__(256) void k_final(
    const float* __restrict__ mlast, const float* __restrict__ ln1b,
    const float* __restrict__ fgate, const float* __restrict__ fcw,
    const float* __restrict__ fcb, float* __restrict__ out)
{
    int idx = blockIdx.x * 256 + threadIdx.x;     // n = b*LL + l
    if (idx >= NSEQ) return;
    int l = idx & (LL - 1);
    int b = idx >> 8;
    float gt = 1.0f / (1.0f + expf(-fgate[l]));
    float fused[DM];
#pragma unroll
    for (int d = 0; d < DM; ++d)
        fused[d] = gt * ln1b[d] + (1.0f - gt) * mlast[(size_t)idx * DM + d];
#pragma unroll
    for (int p = 0; p < PREDN; ++p) {
        float acc = fcb[p];
#pragma unroll
        for (int d = 0; d < DM; ++d) acc += fcw[p * DM + d] * fused[d];
        out[((size_t)b * PREDN + p) * LL + l] = acc;
    }
}

// ---------------------------------------------------------------------------
extern "C" void kernel_launch(void* const* d_in, const int* in_sizes, int n_in,
                              void* d_out, int out_size, void* d_ws, size_t ws_size,
                              hipStream_t stream)
{
    (void)out_size; (void)ws_size;
    enum { IX, IEMBW, IEMBB, IPOS, IAGGW, IAGGB, ISIMW, ILN1G, ILN1B,
           IRMS0, IINP0, ICW0, ICB0, IXP0, IDTW0, IDTB0, IALOG0, ID0, IOUTP0,
           IRMS1, IINP1, ICW1, ICB1, IXP1, IDTW1, IDTB1, IALOG1, ID1, IOUTP1,
           IFW1, IFB1, IFW2, IFB2, ILN2G, ILN2B, IFG, IFCW, IFCB, NIN };
    // Insertion-order flattening of setup_inputs().
    static const int ins_map[NIN] = {
        0,1,2,3,4,5,6,7,8,
        9,10,11,12,13,14,15,16,17,18,
        19,20,21,22,23,24,25,26,27,28,
        29,30,31,32,33,34,35,36,37 };
    // JAX pytree (sorted dict keys) flattening: params first, then x.
    static const int srt_map[NIN] = {
        37, 3, 2, 35, 1, 0, 36, 12, 11,
        23, 21, 18, 17, 24, 20, 19, 15, 16, 22,
        33, 31, 28, 27, 34, 30, 29, 25, 26, 32,
        8, 6, 9, 7, 14, 13, 10, 5, 4 };
    const int* map = (n_in > 0 && in_sizes[0] == BB * TT * LL * INF_) ? ins_map
                                                                      : srt_map;
    const float* P[NIN];
    for (int i = 0; i < NIN; ++i) P[i] = (const float*)d_in[map[i]];

    // Workspace layout (floats).
    float* ws      = (float*)d_ws;
    float* hm      = ws;                                 // NROWS*32
    float* u       = hm   + (size_t)NROWS * DM;          // NROWS*32
    float* xzb     = u    + (size_t)NROWS * DM;          // NROWS*128
    float* xcb     = xzb  + (size_t)NROWS * 2 * DI;      // NROWS*64
    float* dblb    = xcb  + (size_t)NROWS * DI;          // NROWS*34
    float* dtbuf   = dblb + (size_t)NROWS * 34;          // NROWS*64
    float* yb      = dtbuf + (size_t)NROWS * DI;         // NROWS*64
    float* ffn1    = yb   + (size_t)NROWS * DI;          // 512*64
    float* lastrow = ffn1 + (size_t)NSEQ * 2 * DM;       // 512*32
    float* mlast   = lastrow + (size_t)NSEQ * DM;        // 512*32

    // K1: embedding + positional + aggregation, into mamba layout.
    k_embed<<<(BB * TT * LL) / 256, 256, 0, stream>>>(
        P[IX], P[IEMBW], P[IEMBB], P[IPOS], P[IAGGW], P[IAGGB], hm);

    // Two mamba layers: hm += mixer(rms_norm(hm)).
    for (int layer = 0; layer < 2; ++layer) {
        int o = layer * 10;
        const float* rmsw = P[IRMS0 + o];
        const float* inp  = P[IINP0 + o];
        const float* cw   = P[ICW0 + o];
        const float* cb   = P[ICB0 + o];
        const float* xpw  = P[IXP0 + o];
        const float* dtw  = P[IDTW0 + o];
        const float* dtbv = P[IDTB0 + o];
        const float* alog = P[IALOG0 + o];
        const float* dvec = P[ID0 + o];
        const float* outp = P[IOUTP0 + o];

        k_rms<<<NROWS / 8, 256, 0, stream>>>(hm, rmsw, u, NROWS);
        // xz = u @ in_proj^T : M=32768, N=128, K=32  -> 16384 waves
        k_gemm_nt<DM><<<(NROWS / 16) * 8 / 4, 128, 0, stream>>>(
            u, DM, inp, nullptr, nullptr, 0, xzb, 2 * DI, NROWS, 2 * DI, 8, 0);
        k_conv<<<(NROWS * DI) / 256, 256, 0, stream>>>(xzb, cw, cb, xcb);
        // dbl = xc @ x_proj^T : M=32768, N=34 (clamped ragged tile), K=64
        k_gemm_nt<DI><<<(NROWS / 16) * 3 / 4, 128, 0, stream>>>(
            xcb, DI, xpw, nullptr, nullptr, 0, dblb, 34, NROWS, 34, 3, 0);
        k_dt<<<(NROWS * DI) / 256, 256, 0, stream>>>(dblb, dtw, dtbv, dtbuf);
        k_scan<<<NSEQ / 4, 128, 0, stream>>>(xcb, dblb, dtbuf, xzb, alog, dvec, yb);
        // hm += y @ out_proj^T : M=32768, N=32, K=64 (in-place residual)
        k_gemm_nt<DI><<<(NROWS / 16) * 2 / 4, 128, 0, stream>>>(
            yb, DI, outp, nullptr, hm, DM, hm, DM, NROWS, DM, 2, 0);
    }

    // FFN + residual + LN, only on the last timestep rows (t = T-1, strided).
    k_gemm_nt<DM><<<(NSEQ / 16) * 4 / 4, 128, 0, stream>>>(
        hm + (TT - 1) * DM, TT * DM, P[IFW1], P[IFB1], nullptr, 0,
        ffn1, 2 * DM, NSEQ, 2 * DM, 4, 1);
    k_gemm_nt<2 * DM><<<(NSEQ / 16) * 2 / 4, 128, 0, stream>>>(
        ffn1, 2 * DM, P[IFW2], P[IFB2], hm + (TT - 1) * DM, TT * DM,
        lastrow, DM, NSEQ, DM, 2, 0);
    k_ln<<<NSEQ / 8, 256, 0, stream>>>(lastrow, P[ILN2G], P[ILN2B], mlast, NSEQ);

    // Fusion (attn_last == ln1_b exactly, see header comment) + fc + transpose.
    k_final<<<(NSEQ + 255) / 256, 256, 0, stream>>>(
        mlast, P[ILN1B], P[IFG], P[IFCW], P[IFCB], (float*)d_out);
}